// ConvTranspose3d_24721831756211
// MI455X (gfx1250) — compile-verified
//
#include <hip/hip_runtime.h>
#include <hip/hip_bf16.h>

typedef __attribute__((ext_vector_type(2))) float v2f;
typedef __attribute__((ext_vector_type(8))) float v8f;

#define OD 66
#define OSP (66 * 66 * 66)          // 287496 elements per (n, co) plane
#define XSP (32 * 32 * 32)          // 32768 elements per (n, ci) plane

// ---------------------------------------------------------------------------
// Kernel 1: fill entire output with bias[co]. Even-coordinate outputs and the
// output_padding fringe are exactly bias; odd-lattice points get overwritten
// by the WMMA kernel (same stream => ordered).
// ---------------------------------------------------------------------------
__global__ void ct3d_bias_fill(const float* __restrict__ bias,
                               float* __restrict__ out, int total4) {
    int i = blockIdx.x * blockDim.x + threadIdx.x;
    if (i >= total4) return;
    int idx = i * 4;                 // OSP divisible by 4 -> co constant in vec
    int co = (idx / OSP) & 31;
    float b = bias[co];
    float4 v = make_float4(b, b, b, b);
    reinterpret_cast<float4*>(out)[i] = v;
}

// ---------------------------------------------------------------------------
// Kernel 2: implicit-GEMM transposed conv on the odd output lattice.
//   out[n, co, 2j+1] = bias[co] + sum_ci sum_k x[n, ci, j+1-k] * w[ci, co, k]
// M tile: 4 jh x 16 jw = 64 rows, N = 32 (all c_out), K = 27 taps x 64 ci.
// 8 waves: wave = (mt<<1)|nt, mt = jh sub-row, nt = co half.
// ---------------------------------------------------------------------------
__global__ void __launch_bounds__(256)
ct3d_wmma(const float* __restrict__ x, const float* __restrict__ w,
          const float* __restrict__ bias, float* __restrict__ out) {
    __shared__ float As[64 * 66];    // [m row][ci], pad 66: b64-aligned, bank-clean
    __shared__ float Bs[32 * 66];    // [co row][ci]

    const int tid  = threadIdx.x;
    const int wave = tid >> 5;       // wave32
    const int lane = tid & 31;
    const int jw0  = blockIdx.x * 16;        // 0,16,32
    const int jh0  = blockIdx.y * 4;         // 0..32
    const int n    = blockIdx.z / 33;
    const int jd   = blockIdx.z % 33;
    const int mt   = wave >> 1;              // jh sub-tile 0..3
    const int nt   = wave & 1;               // co half 0..1

    // A-staging coordinates for this thread (one M row, 16 ci per tap)
    const int m    = tid & 63;
    const int jh_s = jh0 + (m >> 4);
    const int jw_s = jw0 + (m & 15);
    const int cis  = tid >> 6;               // 0..3, +4 per iteration

    // WMMA fragment LDS offsets (A: lanes 0-15 K={0,1}, 16-31 K={2,3}; B mirrored)
    const int hi       = (lane & 16) ? 2 : 0;
    const int arow_off = (mt * 16 + (lane & 15)) * 66 + hi;
    const int brow_off = (nt * 16 + (lane & 15)) * 66 + hi;

    v8f acc = {};

    for (int kd = 0; kd < 3; ++kd) {
        const int id = jd + 1 - kd;
        const bool dok = (id >= 0) & (id < 32);
        for (int kh = 0; kh < 3; ++kh) {
            const int ih = jh_s + 1 - kh;
            for (int kw = 0; kw < 3; ++kw) {
                const int tap = (kd * 3 + kh) * 3 + kw;
                __syncthreads();             // protect prior-iter LDS reads

                // ---- stage B: 64x32 per-tap weight slice (8 floats/thread)
                #pragma unroll
                for (int it = 0; it < 8; ++it) {
                    int idx = it * 256 + tid;          // 0..2047
                    int ci = idx & 63, co = idx >> 6;  // co 0..31
                    Bs[co * 66 + ci] = w[(ci * 32 + co) * 27 + tap];
                }

                // ---- stage A: 64 rows x 64 ci gathered from x (16/thread)
                const int iw = jw_s + 1 - kw;
                const bool ok = dok & (ih >= 0) & (ih < 32) &
                                (iw >= 0) & (iw < 32) &
                                (jh_s <= 32) & (jw_s <= 32);
                const float* xp =
                    x + (size_t)(((n * 64) * 32 + id) * 32 + ih) * 32 + iw;
                #pragma unroll
                for (int cg = 0; cg < 16; ++cg) {
                    int ci = cg * 4 + cis;
                    As[m * 66 + ci] = ok ? xp[(size_t)ci * XSP] : 0.0f;
                }
                __syncthreads();

                // ---- 16 WMMA K-steps over ci (K=4 each), fp32-exact
                #pragma unroll
                for (int kk = 0; kk < 16; ++kk) {
                    v2f a = *reinterpret_cast<const v2f*>(&As[arow_off + kk * 4]);
                    v2f b = *reinterpret_cast<const v2f*>(&Bs[brow_off + kk * 4]);
                    acc = __builtin_amdgcn_wmma_f32_16x16x4_f32(
                        false, a, false, b, (short)0, acc, false, false);
                }
            }
        }
    }

    // ---- store: C/D layout = VGPR v, lanes 0-15: (M=v, N=lane); 16-31: (M=v+8)
    const int co = nt * 16 + (lane & 15);
    const int jh = jh0 + mt;
    const float bv = bias[co];
    const int mbase = (lane & 16) ? 8 : 0;
    if (jh <= 32) {
        const int od = 2 * jd + 1, oh = 2 * jh + 1;
        float* op = out + (size_t)((n * 32 + co) * OD + od) * (OD * OD) + oh * OD;
        #pragma unroll
        for (int v = 0; v < 8; ++v) {
            int jw = jw0 + mbase + v;        // M row of wave tile = jw offset
            if (jw <= 32) op[2 * jw + 1] = acc[v] + bv;
        }
    }
}

extern "C" void kernel_launch(void* const* d_in, const int* in_sizes, int n_in,
                              void* d_out, int out_size, void* d_ws, size_t ws_size,
                              hipStream_t stream) {
    const float* x    = (const float*)d_in[0];   // (2,64,32,32,32)
    const float* wgt  = (const float*)d_in[1];   // (64,32,3,3,3)
    const float* bias = (const float*)d_in[2];   // (32,)
    float* out = (float*)d_out;                  // (2,32,66,66,66)

    int total4 = out_size / 4;                   // 4,599,936 float4 fills
    ct3d_bias_fill<<<(total4 + 255) / 256, 256, 0, stream>>>(bias, out, total4);

    dim3 grid(3 /*jw tiles*/, 9 /*jh blocks*/, 66 /*n * jd*/);
    ct3d_wmma<<<grid, 256, 0, stream>>>(x, wgt, bias, out);
}